// RNN_72688026517902
// MI455X (gfx1250) — compile-verified
//
#include <hip/hip_runtime.h>

// ---------------- problem constants (match reference) ----------------
#define T_STEPS   512
#define BATCH     256
#define RULE_ST   85
#define N_RULE    20
#define N_IN      105   // RULE_ST + N_RULE
#define N_OUT     33
#define HID       256
#define ALPHA_C   0.2f
#define SIGMA_C   0.05f

typedef __attribute__((ext_vector_type(2))) float v2f;
typedef __attribute__((ext_vector_type(8))) float v8f;
typedef __attribute__((ext_vector_type(4))) int   v4i;

// ---- async global->LDS path (gfx1250), guarded so compile never breaks ----
#if defined(__has_builtin)
#if __has_builtin(__builtin_amdgcn_global_load_async_to_lds_b128) && \
    __has_builtin(__builtin_amdgcn_s_wait_asynccnt)
#define HAVE_ASYNC_LDS 1
#endif
#endif
#ifndef HAVE_ASYNC_LDS
#define HAVE_ASYNC_LDS 0
#endif

#if HAVE_ASYNC_LDS
typedef __attribute__((address_space(1))) v4i* gptr_v4i;   // global int4*
typedef __attribute__((address_space(3))) v4i* lptr_v4i;   // LDS int4*
#endif

__device__ __forceinline__ float softplus_f(float x) {
    // numerically stable: max(x,0) + log1p(exp(-|x|))
    return fmaxf(x, 0.0f) + log1pf(expf(-fabsf(x)));
}

// =====================================================================
// Kernel 1: u[t,b,h] = x[t,b,:]·Wcat[h,:] + b_sens[h] + SIGMA*noise[t,b,h]
// GEMM M=T*B, N=HID, K=105 (padded to 108), f32 WMMA 16x16x4.
// 3 independent accumulator chains to break the WMMA RAW chain.
// =====================================================================
__global__ __launch_bounds__(256)
void u_gemm_kernel(const float* __restrict__ x,
                   const float* __restrict__ noise,
                   const float* __restrict__ W_sens,
                   const float* __restrict__ b_sens,
                   const float* __restrict__ W_rule,
                   float* __restrict__ u_ws)
{
    const int tid  = threadIdx.x;
    const int lane = tid & 31;
    const int li   = lane & 15;
    const int half = lane >> 4;
    const int wave = blockIdx.x * (blockDim.x >> 5) + (tid >> 5);

    const int mtile = wave >> 4;        // 8192 row tiles of flattened T*B
    const int ntile = wave & 15;        // 16 col tiles of HID
    const int mbase = mtile * 16;
    const int nbase = ntile * 16;

    const int arow = mbase + li;        // A-matrix M index = lane 0..15
    const int ncol = nbase + li;        // B-matrix N index

    v8f accs[3] = {};
#pragma unroll
    for (int ks = 0; ks < 27; ++ks) {   // K padded 105 -> 108
        const int k = ks * 4 + half * 2;
        v2f a; a.x = 0.0f; a.y = 0.0f;
        if (k     < N_IN) a.x = x[arow * N_IN + k];
        if (k + 1 < N_IN) a.y = x[arow * N_IN + k + 1];
        // B[k][n] = Wcat[n][k] : first 85 from W_sens, next 20 from W_rule
        v2f b; b.x = 0.0f; b.y = 0.0f;
        if      (k     < RULE_ST) b.x = W_sens[ncol * RULE_ST + k];
        else if (k     < N_IN)    b.x = W_rule[ncol * N_RULE + (k - RULE_ST)];
        if      (k + 1 < RULE_ST) b.y = W_sens[ncol * RULE_ST + k + 1];
        else if (k + 1 < N_IN)    b.y = W_rule[ncol * N_RULE + (k + 1 - RULE_ST)];

        accs[ks % 3] = __builtin_amdgcn_wmma_f32_16x16x4_f32(
                           false, a, false, b, (short)0, accs[ks % 3], false, false);
    }
    const v8f acc = (accs[0] + accs[1]) + accs[2];

    const float bs = b_sens[ncol];
#pragma unroll
    for (int r = 0; r < 8; ++r) {       // D: M = r + 8*half, N = li
        const int row = mbase + r + half * 8;
        const size_t idx = (size_t)row * HID + ncol;
        u_ws[idx] = acc[r] + bs + SIGMA_C * noise[idx];
    }
}

// =====================================================================
// Kernel 2: fused sequential scan + output projection.
// 16 workgroups (one per 16-row batch tile), 608 threads = 19 waves:
//   waves 0..15 : recurrence, wave w owns hidden cols [16w,16w+16)
//                 Wm = W_rec*mask B-fragments live in 128 VGPRs
//   waves 16..18: output projection tiles (N_OUT=33 padded to 48),
//                 overlapped one step behind the recurrence
// h state ping-pongs between two 16x256 LDS buffers; u[t] tiles are
// double-buffered into LDS via async global->LDS DMA when available.
// =====================================================================

#if HAVE_ASYNC_LDS
__device__ __forceinline__ void stage_u_tile(const float* __restrict__ u_ws,
                                             float* lds_dst, int t, int bbase,
                                             int tid, int nthreads)
{
    // u[t, bbase:bbase+16, :] is one contiguous 16*HID-float block
    float* gsrc = const_cast<float*>(&u_ws[((size_t)t * BATCH + bbase) * HID]);
    const int nvec = (16 * HID) / 4;                 // 1024 x b128 transfers
    for (int i = tid; i < nvec; i += nthreads) {
        __builtin_amdgcn_global_load_async_to_lds_b128(
            (gptr_v4i)(gsrc + i * 4),
            (lptr_v4i)(lds_dst + i * 4),
            0, 0);
    }
}
#endif

__global__ __launch_bounds__(608)
void rnn_scan_kernel(const float* __restrict__ u_ws,
                     const float* __restrict__ W_rec,
                     const float* __restrict__ b_rec,
                     const float* __restrict__ maskp,
                     const float* __restrict__ W_out,
                     const float* __restrict__ b_out,
                     float* __restrict__ out)
{
    __shared__ float hbuf[2][16][HID];      // 32 KB, h state ping-pong
#if HAVE_ASYNC_LDS
    __shared__ float ubuf[2][16][HID];      // 32 KB, u[t] DMA double buffer
#endif

    const int tid   = threadIdx.x;
    const int wave  = tid >> 5;
    const int lane  = tid & 31;
    const int li    = lane & 15;
    const int half  = lane >> 4;
    const int bbase = blockIdx.x * 16;

#if HAVE_ASYNC_LDS
    stage_u_tile(u_ws, &ubuf[0][0][0], 0, bbase, tid, (int)blockDim.x);
#endif

    // h0 = 0
    for (int i = tid; i < 16 * HID; i += blockDim.x)
        ((float*)hbuf)[i] = 0.0f;

    v2f   bfrag[64];                    // register-resident B fragments
    float bias_n = 0.0f;
    v8f   h_reg  = {};                  // this wave's 16x16 h tile (C/D layout)

    if (wave < 16) {
        const int n = wave * 16 + li;   // hidden column
#pragma unroll
        for (int ks = 0; ks < 64; ++ks) {
            const int k = ks * 4 + half * 2;
            bfrag[ks].x = W_rec[n * HID + k    ] * maskp[n * HID + k    ];
            bfrag[ks].y = W_rec[n * HID + k + 1] * maskp[n * HID + k + 1];
        }
        bias_n = b_rec[n];
    } else {
        const int o = (wave - 16) * 16 + li;   // output column (pad >=33 with 0)
#pragma unroll
        for (int ks = 0; ks < 64; ++ks) {
            const int k = ks * 4 + half * 2;
            bfrag[ks].x = (o < N_OUT) ? W_out[o * HID + k    ] : 0.0f;
            bfrag[ks].y = (o < N_OUT) ? W_out[o * HID + k + 1] : 0.0f;
        }
        bias_n = (o < N_OUT) ? b_out[o] : 0.0f;
    }

#if HAVE_ASYNC_LDS
    __builtin_amdgcn_s_wait_asynccnt(0);
#endif
    __syncthreads();

    // invariant at top of iter t: hbuf[t&1] holds hs[t-1] (h0 for t=0),
    // and (async path) ubuf[t&1] holds u[t]
    for (int t = 0; t < T_STEPS; ++t) {
        const int p = t & 1;

#if HAVE_ASYNC_LDS
        if (t + 1 < T_STEPS)            // DMA-prefetch next step's u tile
            stage_u_tile(u_ws, &ubuf[1 - p][0][0], t + 1, bbase, tid, (int)blockDim.x);
#endif

        if (wave < 16) {
            const int n = wave * 16 + li;
#if !HAVE_ASYNC_LDS
            __builtin_prefetch(&u_ws[((size_t)t * BATCH + bbase + half * 8) * HID + n], 0, 0);
#endif
            v8f accs[4] = {};           // 4 independent WMMA chains
#pragma unroll
            for (int ks = 0; ks < 64; ++ks) {
                const int k = ks * 4 + half * 2;
                const v2f a = *(const v2f*)&hbuf[p][li][k];   // A: M=li, K=k..k+1
                accs[ks & 3] = __builtin_amdgcn_wmma_f32_16x16x4_f32(
                                   false, a, false, bfrag[ks], (short)0,
                                   accs[ks & 3], false, false);
            }
            const v8f acc = (accs[0] + accs[1]) + (accs[2] + accs[3]);
#pragma unroll
            for (int r = 0; r < 8; ++r) {
                const int m  = r + half * 8;                  // batch row in tile
#if HAVE_ASYNC_LDS
                const float uv = ubuf[p][m][n];
#else
                const float uv = u_ws[((size_t)t * BATCH + bbase + m) * HID + n];
#endif
                const float pre = uv + acc[r] + bias_n;
                const float hn  = ALPHA_C * softplus_f(pre) + (1.0f - ALPHA_C) * h_reg[r];
                h_reg[r] = hn;
                hbuf[1 - p][m][n] = hn;                       // publish hs[t]
            }
        } else if (t > 0) {
            // project hs[t-1] (buffer p) -> out[t-1]
            const int o = (wave - 16) * 16 + li;
            v8f accs[4] = {};
#pragma unroll
            for (int ks = 0; ks < 64; ++ks) {
                const int k = ks * 4 + half * 2;
                const v2f a = *(const v2f*)&hbuf[p][li][k];
                accs[ks & 3] = __builtin_amdgcn_wmma_f32_16x16x4_f32(
                                   false, a, false, bfrag[ks], (short)0,
                                   accs[ks & 3], false, false);
            }
            const v8f acc = (accs[0] + accs[1]) + (accs[2] + accs[3]);
            if (o < N_OUT) {
#pragma unroll
                for (int r = 0; r < 8; ++r) {
                    const int m = r + half * 8;
                    out[((size_t)(t - 1) * BATCH + bbase + m) * N_OUT + o] = acc[r] + bias_n;
                }
            }
        }

#if HAVE_ASYNC_LDS
        __builtin_amdgcn_s_wait_asynccnt(0);   // t+1 tile landed before barrier
#endif
        __syncthreads();
    }

    // epilogue: out[T-1] from hbuf[T&1] (holds hs[T-1])
    if (wave >= 16) {
        const int o = (wave - 16) * 16 + li;
        v8f accs[4] = {};
#pragma unroll
        for (int ks = 0; ks < 64; ++ks) {
            const int k = ks * 4 + half * 2;
            const v2f a = *(const v2f*)&hbuf[T_STEPS & 1][li][k];
            accs[ks & 3] = __builtin_amdgcn_wmma_f32_16x16x4_f32(
                               false, a, false, bfrag[ks], (short)0,
                               accs[ks & 3], false, false);
        }
        const v8f acc = (accs[0] + accs[1]) + (accs[2] + accs[3]);
        if (o < N_OUT) {
#pragma unroll
            for (int r = 0; r < 8; ++r) {
                const int m = r + half * 8;
                out[((size_t)(T_STEPS - 1) * BATCH + bbase + m) * N_OUT + o] = acc[r] + bias_n;
            }
        }
    }
}

// =====================================================================
extern "C" void kernel_launch(void* const* d_in, const int* in_sizes, int n_in,
                              void* d_out, int out_size, void* d_ws, size_t ws_size,
                              hipStream_t stream) {
    const float* x      = (const float*)d_in[0];
    const float* noise  = (const float*)d_in[1];
    const float* W_sens = (const float*)d_in[2];
    const float* b_sens = (const float*)d_in[3];
    const float* W_rule = (const float*)d_in[4];
    const float* W_rec  = (const float*)d_in[5];
    const float* b_rec  = (const float*)d_in[6];
    const float* maskp  = (const float*)d_in[7];
    const float* W_out  = (const float*)d_in[8];
    const float* b_out  = (const float*)d_in[9];
    float* out  = (float*)d_out;
    float* u_ws = (float*)d_ws;   // needs T*B*H*4 = 128 MiB

    // Kernel 1: (T*B/16)=8192 row tiles x 16 col tiles = 131072 wave-tiles,
    // 8 waves per 256-thread block -> 16384 blocks.
    u_gemm_kernel<<<16384, 256, 0, stream>>>(x, noise, W_sens, b_sens, W_rule, u_ws);

    // Kernel 2: one workgroup per 16-row batch tile.
    rnn_scan_kernel<<<BATCH / 16, 608, 0, stream>>>(u_ws, W_rec, b_rec, maskp,
                                                    W_out, b_out, out);
}